// Tract_18769007083560
// MI455X (gfx1250) — compile-verified
//
#include <hip/hip_runtime.h>
#include <cstdint>

// Vocal-tract waveguide sim for MI455X (gfx1250, wave32).
//
// in_audio (B,1,L) f32, reflection (B,45,L) f32, fade (1,1,44) f32 ->
// out (B,1,L) f32.  B=4, L=16384, TRACT_N=44, SIM_STEPS=20 (reference-fixed).
//
// One thread per (b,t) column; 64 sequential scattering steps with the
// 88-float state in registers.  A 256-column reflection tile, transposed to
// [time][k], is staged into LDS with the CDNA5 async global->LDS engine
// (ASYNCcnt / s_wait_asynccnt).  Per step each thread reads its 45
// coefficients from LDS (lane stride = 45 dwords -> conflict-free,
// gcd(45,64)=1; vectorized to ds_load_b128).  Phase-2 "last" values go to a
// workspace; a second kernel applies the forward scatter as a deterministic
// gather (no float atomics).

#define TRACT   44
#define NR      45            // reflection rows (TRACT_N + 1)
#define NSTEPS  64            // TRACT_N + SIM_STEPS
#define PH2     20            // SIM_STEPS
#define TILE    256
#define TW      (TILE + 64)   // tile time-width incl. halo (320)

constexpr int kL        = 16384;        // time length (reference-fixed)
constexpr int kTilesPB  = kL / TILE;    // 64 tiles per batch

// ---- CDNA5 async global -> LDS copy (b32) -------------------------------
typedef __attribute__((address_space(1))) int g_int_t;   // global
typedef __attribute__((address_space(3))) int l_int_t;   // LDS

__device__ __forceinline__ void async_ld_b32(const float* g, float* l) {
#if __has_builtin(__builtin_amdgcn_global_load_async_to_lds_b32)
  __builtin_amdgcn_global_load_async_to_lds_b32(
      (g_int_t*)(uintptr_t)g,
      (l_int_t*)(uint32_t)(uintptr_t)l,
      /*offset=*/0, /*cpol=*/0);
#else
  unsigned long long ga = (unsigned long long)(uintptr_t)g;
  unsigned int lo = (unsigned int)(uintptr_t)l;     // generic low 32 bits == LDS offset
  asm volatile("global_load_async_to_lds_b32 %0, %1, off"
               :: "v"(lo), "v"(ga) : "memory");
#endif
}

__device__ __forceinline__ void wait_async0() {
#if __has_builtin(__builtin_amdgcn_s_wait_asynccnt)
  __builtin_amdgcn_s_wait_asynccnt(0);
#else
  asm volatile("s_wait_asynccnt 0" ::: "memory");
#endif
}

// ---- Kernel 1: per-column 64-step simulation ----------------------------
__global__ __launch_bounds__(TILE)
void tract_sim_kernel(const float* __restrict__ audio,
                      const float* __restrict__ refl,
                      const float* __restrict__ fade,
                      float* __restrict__ out,
                      float* __restrict__ ws,
                      int BL) {
  __shared__ float lds_r[TW * NR];          // [time][k], 57,600 bytes

  const int tid      = threadIdx.x;
  const int b        = blockIdx.x >> 6;                 // / kTilesPB (64)
  const int t0       = (blockIdx.x & (kTilesPB - 1)) * TILE;
  const float* rbase = refl + (size_t)b * NR * kL;

  // --- stage reflection tile into LDS (transposed), async engine ---
  if (t0 + TW > kL) {
    // tail tile: pre-zero so times >= L read as the reference's zero padding
    for (int idx = tid; idx < TW * NR; idx += TILE) lds_r[idx] = 0.0f;
    __syncthreads();
  }
  for (int idx = tid; idx < NR * TW; idx += TILE) {
    const int k  = idx / TW;
    const int tt = idx - k * TW;
    const int t  = t0 + tt;
    if (t < kL) async_ld_b32(rbase + (size_t)k * kL + t, &lds_r[tt * NR + k]);
  }
  wait_async0();
  __syncthreads();

  // --- sigmoid(fade) hoisted into SGPRs (uniform across the wave) ---
  float fs[TRACT];
#pragma unroll
  for (int k = 0; k < TRACT; ++k) {
    const float s = 1.0f / (1.0f + expf(-fade[k]));
    fs[k] = __int_as_float(__builtin_amdgcn_readfirstlane(__float_as_int(s)));
  }

  // --- per-column state in registers ---
  const int gid = b * kL + t0 + tid;
  float tr[TRACT], tl[TRACT];
#pragma unroll
  for (int k = 0; k < TRACT; ++k) { tr[k] = 0.0f; tl[k] = 0.0f; }
  tr[0] = audio[gid];

#pragma unroll 1
  for (int i = 0; i < NSTEPS; ++i) {
    const float* rp = &lds_r[(tid + i) * NR];
    float rr[NR];
#pragma unroll
    for (int k = 0; k < NR; ++k) rr[k] = rp[k];   // -> ds_load_b128 x11 + b32

    float prev = 0.0f;                            // tr_old[k-1] (0 at k==0)
#pragma unroll
    for (int k = 0; k < TRACT; ++k) {
      const float ctr = tr[k];
      const float ctl = tl[k];
      const float sl  = (k == TRACT - 1) ? 0.0f : tl[k + 1];
      const float nr  = prev - (prev + ctl) * rr[k];       // fma form
      const float nl  = sl   + (sl   + ctr) * rr[k + 1];
      tr[k] = nr * fs[k];
      tl[k] = nl * fs[k];
      prev  = ctr;
    }

    if (i == TRACT - 1) {
      out[gid] = tr[TRACT - 1];                   // end of phase 1
    } else if (i >= TRACT) {
      ws[(size_t)(i - TRACT) * BL + gid] = tr[TRACT - 1];  // phase-2 "last"
    }
  }
}

// ---- Kernel 2: deterministic scatter-as-gather --------------------------
__global__ __launch_bounds__(256)
void tract_gather_kernel(float* __restrict__ out,
                         const float* __restrict__ ws,
                         int BL) {
  const int i = blockIdx.x * blockDim.x + threadIdx.x;
  if (i >= BL) return;
  const int t = i & (kL - 1);                     // time within batch
  float acc = out[i];
#pragma unroll
  for (int r = 0; r < PH2; ++r) {
    if (t - r - 1 >= 0) acc += ws[(size_t)r * BL + (i - r - 1)];
  }
  out[i] = acc;
}

// ---- host launch --------------------------------------------------------
extern "C" void kernel_launch(void* const* d_in, const int* in_sizes, int n_in,
                              void* d_out, int out_size, void* d_ws, size_t ws_size,
                              hipStream_t stream) {
  const float* audio = (const float*)d_in[0];   // (B,1,L)
  const float* refl  = (const float*)d_in[1];   // (B,45,L)
  const float* fade  = (const float*)d_in[2];   // (1,1,44)
  float* out = (float*)d_out;
  float* ws  = (float*)d_ws;                    // needs PH2*B*L*4 = 5 MB

  const int BL = out_size;                      // B * L
  const int B  = BL / kL;
  (void)in_sizes; (void)n_in; (void)ws_size;

  const int blocks1 = B * kTilesPB;
  tract_sim_kernel<<<blocks1, TILE, 0, stream>>>(audio, refl, fade, out, ws, BL);

  const int blocks2 = (BL + 255) / 256;
  tract_gather_kernel<<<blocks2, 256, 0, stream>>>(out, ws, BL);
}